// DFTB_Layer_3667902071186
// MI455X (gfx1250) — compile-verified
//
#include <hip/hip_runtime.h>
#include <math.h>

#define NGEO 1024
#define BD   64
#define BB   (BD*BD)          // 4096
#define LDG  66               // GEMM LDS stride: 8B-aligned pairs + conflict-free fragments
#define LDE  65               // Jacobi LDS stride: conflict-free column access
#define RROT 1000000
#define EREPN 2000000
#define NEL  7

// ---- d_out layout (floats) ----
#define OUT_H     0
#define OUT_DQ    (NGEO*BB)                 // 4194304
#define OUT_EREP  (OUT_DQ + NGEO*BD)        // 4259840
#define OUT_EORB  (OUT_EREP + NGEO)         // 4260864
#define OUT_RHO   (OUT_EORB + NGEO*BD)      // 4326400
#define OUT_EELEC (OUT_RHO + NGEO*BB)       // 8520704
#define OUT_EREF  (OUT_EELEC + NGEO)        // 8521728

// ---- d_ws layout (floats) ----
#define WS_ROT    0                         // 2 + 9*R, padded
#define WS_ROTLEN 9000064
#define WS_FOCKP  WS_ROTLEN
#define WS_C      (WS_FOCKP + NGEO*BB)
#define WS_ENER2  (WS_C + NGEO*BB)

typedef float v2f __attribute__((ext_vector_type(2)));
typedef float v8f __attribute__((ext_vector_type(8)));

// ---------- WMMA helpers: 64x64x64 f32 GEMM tile via V_WMMA_F32_16X16X4_F32 ----------
// A,B are 64x64 with row strides lda/ldb (LDS, padded). transA: use A^T; transB: use B^T.
__device__ __forceinline__ v8f wmma_tile64(const float* A, int lda,
                                           const float* B, int ldb,
                                           int m0, int n0, bool transA, bool transB) {
    const int lane = threadIdx.x & 31;
    const int half = lane >> 4;       // 0 or 1
    const int l15  = lane & 15;
    v8f acc = {};
    #pragma unroll
    for (int k0 = 0; k0 < BD; k0 += 4) {
        const int ka = k0 + 2 * half; // this lane's K start (0/2 within group of 4)
        v2f a, b;
        if (!transA) {
            a.x = A[(m0 + l15) * lda + ka];
            a.y = A[(m0 + l15) * lda + ka + 1];
        } else {
            a.x = A[ka * lda + (m0 + l15)];
            a.y = A[(ka + 1) * lda + (m0 + l15)];
        }
        if (!transB) {
            b.x = B[ka * ldb + (n0 + l15)];
            b.y = B[(ka + 1) * ldb + (n0 + l15)];
        } else {
            b.x = B[(n0 + l15) * ldb + ka];
            b.y = B[(n0 + l15) * ldb + ka + 1];
        }
        acc = __builtin_amdgcn_wmma_f32_16x16x4_f32(false, a, false, b,
                                                    (short)0, acc, false, false);
    }
    return acc;
}

__device__ __forceinline__ void wmma_store_tile(float* D, int ldd, int m0, int n0,
                                                v8f acc, float scale) {
    const int lane = threadIdx.x & 31;
    const int half = lane >> 4;
    const int l15  = lane & 15;
    #pragma unroll
    for (int v = 0; v < 8; ++v)
        D[(m0 + v + 8 * half) * ldd + (n0 + l15)] = scale * acc[v];
}

// ---------- Kernel A: rotation stage  rot[r,9] = T[r,9,3] @ net[gather_rot[r,3]] ----------
__global__ void k_rot(const float* __restrict__ net, const float* __restrict__ T,
                      const int* __restrict__ gidx, float* __restrict__ rot_out) {
    int r = blockIdx.x * blockDim.x + threadIdx.x;
    if (r == 0) { rot_out[0] = 0.0f; rot_out[1] = 1.0f; }
    if (r >= RROT) return;
    float v0 = net[gidx[3 * r + 0]];
    float v1 = net[gidx[3 * r + 1]];
    float v2 = net[gidx[3 * r + 2]];
    const float* t = T + (size_t)r * 27;
    float* o = rot_out + 2 + (size_t)r * 9;
    #pragma unroll
    for (int j = 0; j < 9; ++j)
        o[j] = t[3 * j] * v0 + t[3 * j + 1] * v1 + t[3 * j + 2] * v2;
}

// ---------- Kernel B: H gather ----------
__global__ void k_hgather(const float* __restrict__ rot_out,
                          const int* __restrict__ goper, float* __restrict__ H) {
    int i0 = (blockIdx.x * blockDim.x + threadIdx.x) * 4;
    #pragma unroll
    for (int k = 0; k < 4; ++k) {
        int i = i0 + k;
        if (i < NGEO * BB) H[i] = rot_out[goper[i]];
    }
}

// ---------- Kernel I: init Erep=0, Eref ----------
__global__ void k_init(const float* __restrict__ zc, const float* __restrict__ rv,
                       float* __restrict__ out) {
    int t = blockIdx.x * blockDim.x + threadIdx.x;
    if (t < NGEO) out[OUT_EREP + t] = 0.0f;
    if (t == 0) {
        float e = 0.0f;
        #pragma unroll
        for (int i = 0; i < NEL; ++i) e += zc[i] * rv[i];
        out[OUT_EREF] = e;
    }
}

// ---------- Kernel C: Erep segment sum (sorted segments, run-length batched atomics) ----------
#define ECH 8
__global__ void k_erep(const float* __restrict__ net, const int* __restrict__ grep,
                       const int* __restrict__ seg, float* __restrict__ erep) {
    int base = (blockIdx.x * blockDim.x + threadIdx.x) * ECH;
    if (base >= EREPN) return;
    int end = min(base + ECH, EREPN);
    int cur = seg[base];
    float acc = net[grep[base]];
    for (int e = base + 1; e < end; ++e) {
        int s = seg[e];
        float v = net[grep[e]];
        if (s == cur) acc += v;
        else { atomicAdd(&erep[cur], acc); cur = s; acc = v; }
    }
    atomicAdd(&erep[cur], acc);
}

// ---------- Kernel D: per-geometry dQ, ep, F, fockp (WMMA) ----------
__global__ void __launch_bounds__(256)
k_fock(const float* __restrict__ S, const float* __restrict__ G,
       const float* __restrict__ rho, const float* __restrict__ qn,
       const float* __restrict__ phiS, float* __restrict__ out,
       float* __restrict__ fockp, float* __restrict__ ener2) {
    __shared__ float Fsh[BD * LDG];
    __shared__ float Psh[BD * LDG];
    __shared__ float Tsh[BD * LDG];
    __shared__ float dQ[BD], ep[BD], red[256];
    const int g = blockIdx.x, t = threadIdx.x;
    const size_t gb = (size_t)g * BB;

    // load phiS (padded LDS)
    #pragma unroll
    for (int k = 0; k < 16; ++k) {
        int idx = t * 16 + k, i = idx >> 6, j = idx & 63;
        Psh[i * LDG + j] = phiS[gb + idx];
    }

    // dQ[i] = qn[i] - sum_j rho[i][j]*S[i][j]   (4 threads per row)
    {
        int i = t >> 2, p = t & 3;
        float part = 0.0f;
        for (int j = p * 16; j < p * 16 + 16; ++j)
            part += rho[gb + i * BD + j] * S[gb + i * BD + j];
        red[t] = part;
    }
    __syncthreads();
    if (t < BD) {
        float s = red[t * 4] + red[t * 4 + 1] + red[t * 4 + 2] + red[t * 4 + 3];
        float d = qn[(size_t)g * BD + t] - s;
        dQ[t] = d;
        out[OUT_DQ + (size_t)g * BD + t] = d;
    }
    __syncthreads();
    // ep = G @ dQ
    {
        int i = t >> 2, p = t & 3;
        float part = 0.0f;
        for (int j = p * 16; j < p * 16 + 16; ++j)
            part += G[gb + i * BD + j] * dQ[j];
        red[t] = part;
    }
    __syncthreads();
    if (t < BD) ep[t] = red[t * 4] + red[t * 4 + 1] + red[t * 4 + 2] + red[t * 4 + 3];
    __syncthreads();
    // ener2 = 0.5 * dQ^T ep
    if (t == 0) {
        float e = 0.0f;
        for (int i = 0; i < BD; ++i) e += dQ[i] * ep[i];
        ener2[g] = 0.5f * e;
    }
    // F = H - 0.5*S*(ep_i + ep_j)
    #pragma unroll
    for (int k = 0; k < 16; ++k) {
        int idx = t * 16 + k, i = idx >> 6, j = idx & 63;
        Fsh[i * LDG + j] = out[OUT_H + gb + idx] - 0.5f * S[gb + idx] * (ep[i] + ep[j]);
    }
    __syncthreads();

    // tmp = F @ phiS   (WMMA, 8 waves x 2 tiles)
    const int wave = t >> 5;
    for (int ti = wave * 2; ti < wave * 2 + 2; ++ti) {
        int m0 = (ti >> 2) * 16, n0 = (ti & 3) * 16;
        v8f acc = wmma_tile64(Fsh, LDG, Psh, LDG, m0, n0, false, false);
        wmma_store_tile(Tsh, LDG, m0, n0, acc, 1.0f);
    }
    __syncthreads();
    // fockp = phiS^T @ tmp
    for (int ti = wave * 2; ti < wave * 2 + 2; ++ti) {
        int m0 = (ti >> 2) * 16, n0 = (ti & 3) * 16;
        v8f acc = wmma_tile64(Psh, LDG, Tsh, LDG, m0, n0, true, false);
        wmma_store_tile(fockp + gb, BD, m0, n0, acc, 1.0f);
    }
}

// ---------- Kernel E: batched Jacobi eigensolver (64x64 symmetric, padded LDS) ----------
#define NSWEEP 10
__global__ void __launch_bounds__(256)
k_eigh(const float* __restrict__ fockp, float* __restrict__ out, float* __restrict__ Cws) {
    __shared__ float A[BD * LDE];
    __shared__ float V[BD * LDE];
    __shared__ float csh[32], ssh[32];
    __shared__ int   psh[32], qsh[32], flg[32];
    __shared__ float d[BD];
    const int g = blockIdx.x, t = threadIdx.x;
    const size_t gb = (size_t)g * BB;

    #pragma unroll
    for (int k = 0; k < 16; ++k) {
        int idx = t * 16 + k, i = idx >> 6, j = idx & 63;
        A[i * LDE + j] = fockp[gb + idx];
        V[i * LDE + j] = (i == j) ? 1.0f : 0.0f;
    }
    __syncthreads();

    for (int sw = 0; sw < NSWEEP; ++sw) {
        for (int r = 0; r < 63; ++r) {
            if (t < 32) {
                int p, q;
                if (t == 0) { p = 63; q = r % 63; }
                else { p = (r + t) % 63; q = (r + 63 - t) % 63; }
                if (p > q) { int tmp = p; p = q; q = tmp; }
                float app = A[p * LDE + p], aqq = A[q * LDE + q], apq = A[p * LDE + q];
                float c = 1.0f, s = 0.0f;
                if (fabsf(apq) > 1e-12f) {
                    float tau = (aqq - app) / (2.0f * apq);
                    float tt = (tau >= 0.0f ? 1.0f : -1.0f) /
                               (fabsf(tau) + sqrtf(1.0f + tau * tau));
                    c = 1.0f / sqrtf(1.0f + tt * tt);
                    s = tt * c;
                }
                csh[t] = c; ssh[t] = s; psh[t] = p; qsh[t] = q;
            }
            __syncthreads();
            // row update: rows p,q across all columns (32 pairs x 64 cols)
            #pragma unroll
            for (int it = 0; it < 8; ++it) {
                int w = t + 256 * it, pr = w >> 6, j = w & 63;
                int p = psh[pr], q = qsh[pr];
                float c = csh[pr], s = ssh[pr];
                float ap = A[p * LDE + j], aq = A[q * LDE + j];
                A[p * LDE + j] = c * ap - s * aq;
                A[q * LDE + j] = s * ap + c * aq;
            }
            __syncthreads();
            // column update on A and V (stride 65 -> bank-conflict-free columns)
            #pragma unroll
            for (int it = 0; it < 8; ++it) {
                int w = t + 256 * it, pr = w >> 6, i = w & 63;
                int p = psh[pr], q = qsh[pr];
                float c = csh[pr], s = ssh[pr];
                float ap = A[i * LDE + p], aq = A[i * LDE + q];
                A[i * LDE + p] = c * ap - s * aq;
                A[i * LDE + q] = s * ap + c * aq;
                float vp = V[i * LDE + p], vq = V[i * LDE + q];
                V[i * LDE + p] = c * vp - s * vq;
                V[i * LDE + q] = s * vp + c * vq;
            }
            __syncthreads();
        }
    }
    if (t < BD) d[t] = A[t * LDE + t];
    __syncthreads();
    // odd-even transposition sort ascending, permuting eigenvector columns
    for (int ph = 0; ph < BD; ++ph) {
        if (t < 32) {
            int i = 2 * t + (ph & 1);
            int doswap = 0;
            if (i + 1 < BD && d[i] > d[i + 1]) {
                float tmp = d[i]; d[i] = d[i + 1]; d[i + 1] = tmp; doswap = 1;
            }
            flg[t] = doswap;
        }
        __syncthreads();
        #pragma unroll
        for (int it = 0; it < 8; ++it) {
            int w = t + 256 * it, pr = w >> 6, row = w & 63;
            if (flg[pr]) {
                int i = 2 * pr + (ph & 1);
                float a = V[row * LDE + i], b = V[row * LDE + i + 1];
                V[row * LDE + i] = b; V[row * LDE + i + 1] = a;
            }
        }
        __syncthreads();
    }
    if (t < BD) out[OUT_EORB + (size_t)g * BD + t] = d[t];
    #pragma unroll
    for (int k = 0; k < 16; ++k) {
        int idx = t * 16 + k, i = idx >> 6, j = idx & 63;
        Cws[gb + idx] = V[i * LDE + j];
    }
}

// ---------- Kernel F: orb, rho_out, Eelec (WMMA) ----------
__global__ void __launch_bounds__(256)
k_rho(const float* __restrict__ phiS, const float* __restrict__ Cws,
      const float* __restrict__ mask, const float* __restrict__ ener2,
      float* __restrict__ out) {
    __shared__ float Psh[BD * LDG];
    __shared__ float Csh[BD * LDG];
    __shared__ float Osh[BD * LDG];
    __shared__ float red[256];
    const int g = blockIdx.x, t = threadIdx.x;
    const size_t gb = (size_t)g * BB;

    #pragma unroll
    for (int k = 0; k < 16; ++k) {
        int idx = t * 16 + k, i = idx >> 6, j = idx & 63;
        Psh[i * LDG + j] = phiS[gb + idx];
        Csh[i * LDG + j] = Cws[gb + idx];
    }
    __syncthreads();
    const int wave = t >> 5;
    // orb = phiS @ C
    for (int ti = wave * 2; ti < wave * 2 + 2; ++ti) {
        int m0 = (ti >> 2) * 16, n0 = (ti & 3) * 16;
        v8f acc = wmma_tile64(Psh, LDG, Csh, LDG, m0, n0, false, false);
        wmma_store_tile(Osh, LDG, m0, n0, acc, 1.0f);
    }
    __syncthreads();
    // orb_filled = mask * orb
    #pragma unroll
    for (int k = 0; k < 16; ++k) {
        int idx = t * 16 + k, i = idx >> 6, j = idx & 63;
        Osh[i * LDG + j] *= mask[gb + idx];
    }
    __syncthreads();
    // rho = 2 * orbf @ orbf^T, fused ener1 = sum(rho*H)
    const int lane = t & 31, half = lane >> 4, l15 = lane & 15;
    float part = 0.0f;
    for (int ti = wave * 2; ti < wave * 2 + 2; ++ti) {
        int m0 = (ti >> 2) * 16, n0 = (ti & 3) * 16;
        v8f acc = wmma_tile64(Osh, LDG, Osh, LDG, m0, n0, false, true);
        #pragma unroll
        for (int v = 0; v < 8; ++v) {
            int row = m0 + v + 8 * half, col = n0 + l15;
            float val = 2.0f * acc[v];
            out[OUT_RHO + gb + row * BD + col] = val;
            part += val * out[OUT_H + gb + row * BD + col];
        }
    }
    red[t] = part;
    __syncthreads();
    if (t == 0) {
        float e = 0.0f;
        for (int i = 0; i < 256; ++i) e += red[i];
        out[OUT_EELEC + g] = e + ener2[g];
    }
}

extern "C" void kernel_launch(void* const* d_in, const int* in_sizes, int n_in,
                              void* d_out, int out_size, void* d_ws, size_t ws_size,
                              hipStream_t stream) {
    const float* net   = (const float*)d_in[0];
    const float* rotT  = (const float*)d_in[1];
    const float* S     = (const float*)d_in[2];
    const float* G     = (const float*)d_in[3];
    const float* rho   = (const float*)d_in[4];
    const float* qn    = (const float*)d_in[5];
    const float* mask  = (const float*)d_in[6];
    const float* phiS  = (const float*)d_in[7];
    const float* zc    = (const float*)d_in[8];
    const float* rv    = (const float*)d_in[9];
    const int* grot    = (const int*)d_in[10];
    const int* goper   = (const int*)d_in[11];
    const int* grep    = (const int*)d_in[12];
    const int* seg     = (const int*)d_in[13];
    float* out = (float*)d_out;
    float* ws  = (float*)d_ws;

    float* rot_out = ws + WS_ROT;
    float* fockp   = ws + WS_FOCKP;
    float* Cws     = ws + WS_C;
    float* ener2   = ws + WS_ENER2;

    k_rot<<<(RROT + 255) / 256, 256, 0, stream>>>(net, rotT, grot, rot_out);
    k_hgather<<<(NGEO * BB + 1023) / 1024, 256, 0, stream>>>(rot_out, goper, out + OUT_H);
    k_init<<<(NGEO + 255) / 256, 256, 0, stream>>>(zc, rv, out);
    k_erep<<<((EREPN + ECH - 1) / ECH + 255) / 256, 256, 0, stream>>>(
        net, grep, seg, out + OUT_EREP);
    k_fock<<<NGEO, 256, 0, stream>>>(S, G, rho, qn, phiS, out, fockp, ener2);
    k_eigh<<<NGEO, 256, 0, stream>>>(fockp, out, Cws);
    k_rho<<<NGEO, 256, 0, stream>>>(phiS, Cws, mask, ener2, out);
}